// GraphAttentionLayer_53953379173308
// MI455X (gfx1250) — compile-verified
//
#include <hip/hip_runtime.h>

// GAT layer fused kernel for gfx1250 (MI455X).
// Output = [out (4*2048*256 f32)] ++ [att (32*2048*2048 f32)], concatenated.
// HBM-bound on the 536MB att store stream -> NT-hinted b128 stores; adj/x kept
// L2-resident (RT); agg matmul on v_wmma_f32_16x16x4_f32.

typedef __attribute__((ext_vector_type(2))) float v2f;
typedef __attribute__((ext_vector_type(4))) float f32x4;
typedef __attribute__((ext_vector_type(8))) float v8f;
typedef __attribute__((ext_vector_type(4))) int   i32x4;

#define B_    4
#define N_    2048
#define H_    8
#define DIN   32
#define DOUT  32
#define XROW  (H_*DIN)   // 256 floats per x row
#define CH    268        // padded LDS row stride (floats); 268*4B keeps 16B align + bank spread

__device__ __forceinline__ float leaky_mask(float xi, float xj, int a) {
    float s = xi + xj;
    s = s > 0.0f ? s : 0.01f * s;      // leaky_relu, slope 0.01
    return a ? s : -1.0e9f;
}

__global__ __launch_bounds__(256)
void gat_fused(const float* __restrict__ x, const int* __restrict__ adj,
               const float* __restrict__ att_w, const float* __restrict__ fc_w,
               const float* __restrict__ fc_b, float* __restrict__ out,
               float* __restrict__ att_out)
{
    __shared__ float aw[64];
    __shared__ float xi_s[16];
    __shared__ alignas(16) float xj_s[N_];
    __shared__ alignas(16) float att_c[16 * CH];
    __shared__ float red[16][17];
    __shared__ float rowmax[16];
    __shared__ float rowsum[16];
    __shared__ float agg[16 * DIN];

    const int tid = threadIdx.x;
    const int g   = blockIdx.y;          // 0..31 (= bi*H + h)
    const int i0  = blockIdx.x * 16;     // row tile start
    const int bi  = g >> 3;
    const int h   = g & 7;

    // xh[g] row j lives at xg + j*XROW (contiguous 32 floats)
    const float* xg = x + (size_t)bi * N_ * XROW + h * DIN;

    if (tid < 64) aw[tid] = att_w[tid];
    for (int idx = tid; idx < 16 * DIN; idx += 256) agg[idx] = 0.0f;
    __syncthreads();

    // xj for the whole group; xi for this tile's 16 rows
    for (int j = tid; j < N_; j += 256) {
        const float* xr = xg + (size_t)j * XROW;
        float s = 0.0f;
        #pragma unroll
        for (int d = 0; d < DIN; ++d) s += xr[d] * aw[DIN + d];
        xj_s[j] = s;
    }
    if (tid < 16) {
        const float* xr = xg + (size_t)(i0 + tid) * XROW;
        float s = 0.0f;
        #pragma unroll
        for (int d = 0; d < DIN; ++d) s += xr[d] * aw[d];
        xi_s[tid] = s;
    }
    __syncthreads();

    const int r = tid >> 4;      // row within tile (0..15), 16 threads per row
    const int l = tid & 15;
    // faithful reference mismatch: adjacency comes from batch (g % 4)
    const int* arow = adj + ((size_t)(g & 3) * N_ + (i0 + r)) * N_;
    const float xir = xi_s[r];

    // ---- pass A: row max of masked leaky_relu scores (b128 loads, coalesced) ----
    float mloc = -3.0e38f;
    for (int jb = l * 4; jb < N_; jb += 64) {
        i32x4 a4  = *(const i32x4*)(arow + jb);
        f32x4 xj4 = *(const f32x4*)(xj_s + jb);
        mloc = fmaxf(mloc, leaky_mask(xir, xj4.x, a4.x));
        mloc = fmaxf(mloc, leaky_mask(xir, xj4.y, a4.y));
        mloc = fmaxf(mloc, leaky_mask(xir, xj4.z, a4.z));
        mloc = fmaxf(mloc, leaky_mask(xir, xj4.w, a4.w));
    }
    red[r][l] = mloc;
    __syncthreads();
    if (tid < 16) {
        float mm = red[tid][0];
        #pragma unroll
        for (int k = 1; k < 16; ++k) mm = fmaxf(mm, red[tid][k]);
        rowmax[tid] = mm;
    }
    __syncthreads();

    // ---- pass B: sum of exp ----
    const float xmax = rowmax[r];
    float sloc = 0.0f;
    for (int jb = l * 4; jb < N_; jb += 64) {
        i32x4 a4  = *(const i32x4*)(arow + jb);
        f32x4 xj4 = *(const f32x4*)(xj_s + jb);
        sloc += __expf(leaky_mask(xir, xj4.x, a4.x) - xmax);
        sloc += __expf(leaky_mask(xir, xj4.y, a4.y) - xmax);
        sloc += __expf(leaky_mask(xir, xj4.z, a4.z) - xmax);
        sloc += __expf(leaky_mask(xir, xj4.w, a4.w) - xmax);
    }
    red[r][l] = sloc;
    __syncthreads();
    if (tid < 16) {
        float ss = 0.0f;
        #pragma unroll
        for (int k = 0; k < 16; ++k) ss += red[tid][k];
        rowsum[tid] = ss;
    }
    __syncthreads();
    const float inv = 1.0f / rowsum[r];

    // ---- pass C: chunks of 256 cols: write att (NT b128 global + LDS), WMMA agg ----
    const int wave = tid >> 5;           // 0..7, each owns a disjoint K-subrange
    const int lane = tid & 31;
    const int m16  = lane & 15;
    const int koff = (lane >> 4) << 1;   // lanes 0-15 -> K+0/K+1, lanes 16-31 -> K+2/K+3
    v8f acc0 = {};                       // D columns 0..15
    v8f acc1 = {};                       // D columns 16..31

    float* attg = att_out + ((size_t)g * N_ + (i0 + r)) * N_;

    for (int c0 = 0; c0 < N_; c0 += 256) {
        #pragma unroll
        for (int jb = l * 4; jb < 256; jb += 64) {
            int j = c0 + jb;
            i32x4 a4  = *(const i32x4*)(arow + j);
            f32x4 xj4 = *(const f32x4*)(xj_s + j);
            f32x4 e4;
            e4.x = __expf(leaky_mask(xir, xj4.x, a4.x) - xmax) * inv;
            e4.y = __expf(leaky_mask(xir, xj4.y, a4.y) - xmax) * inv;
            e4.z = __expf(leaky_mask(xir, xj4.z, a4.z) - xmax) * inv;
            e4.w = __expf(leaky_mask(xir, xj4.w, a4.w) - xmax) * inv;
            *(f32x4*)(att_c + r * CH + jb) = e4;
            // write-once 536MB stream: non-temporal so adj/x stay L2-resident
            __builtin_nontemporal_store(e4, (f32x4*)(attg + j));
        }
        __syncthreads();

        // A = att tile (16 x 4 f32), B = xh rows (4 x 16 f32), C += A*B
        #pragma unroll
        for (int ks = 0; ks < 8; ++ks) {
            int kk = wave * 32 + ks * 4;           // K offset within chunk
            int kg = c0 + kk + koff;               // global K row
            v2f A, B0, B1;
            A.x = att_c[m16 * CH + kk + koff];
            A.y = att_c[m16 * CH + kk + koff + 1];
            const float* r0 = xg + (size_t)kg * XROW;   // L2-resident
            B0.x = r0[m16];           B0.y = r0[XROW + m16];
            B1.x = r0[m16 + 16];      B1.y = r0[XROW + m16 + 16];
            acc0 = __builtin_amdgcn_wmma_f32_16x16x4_f32(false, A, false, B0,
                                                         (short)0, acc0, false, false);
            acc1 = __builtin_amdgcn_wmma_f32_16x16x4_f32(false, A, false, B1,
                                                         (short)0, acc1, false, false);
        }
        __syncthreads();   // protect att_c before next chunk overwrites it
    }

    // ---- reduce 8 wave-partials into agg[16][32] via LDS float atomics ----
    {
        const int rbase = (lane >> 4) ? 8 : 0;     // C/D layout: VGPR v -> M = v (+8 hi half)
        #pragma unroll
        for (int v = 0; v < 8; ++v) {
            atomicAdd(&agg[(v + rbase) * DIN + m16],       acc0[v]);
            atomicAdd(&agg[(v + rbase) * DIN + 16 + m16],  acc1[v]);
        }
    }
    __syncthreads();

    // ---- FC: out = relu(agg @ fc_w^T + fc_b), scatter back to (b, n, H*DOUT) ----
    for (int idx = tid; idx < 16 * DOUT; idx += 256) {
        int rr = idx >> 5;
        int oc = idx & 31;
        float acc = fc_b[oc];
        #pragma unroll
        for (int d = 0; d < DIN; ++d) acc += agg[rr * DIN + d] * fc_w[oc * DIN + d];
        out[((size_t)bi * N_ + i0 + rr) * (H_ * DOUT) + h * DOUT + oc] = fmaxf(acc, 0.0f);
    }
}

extern "C" void kernel_launch(void* const* d_in, const int* in_sizes, int n_in,
                              void* d_out, int out_size, void* d_ws, size_t ws_size,
                              hipStream_t stream) {
    const float* x     = (const float*)d_in[0];
    const int*   adj   = (const int*)d_in[1];
    const float* att_w = (const float*)d_in[2];
    const float* fc_w  = (const float*)d_in[3];
    const float* fc_b  = (const float*)d_in[4];

    float* out = (float*)d_out;
    float* att = out + (size_t)B_ * N_ * (H_ * DOUT);   // second tuple element

    dim3 grid(N_ / 16, B_ * H_);   // 128 row-tiles x 32 groups
    gat_fused<<<grid, dim3(256), 0, stream>>>(x, adj, att_w, fc_w, fc_b, out, att);
}